// PreferenceEncoder_48352741819110
// MI455X (gfx1250) — compile-verified
//
#include <hip/hip_runtime.h>
#include <hip/hip_bf16.h>

#define N_USERS  100000
#define N_ITEMS  200000
#define N_NODES  (N_USERS + N_ITEMS)      // 300000
#define DIM      128
#define N_EDGES  600000
#define N_LAYERS 3
#define TOTAL    (N_NODES * DIM)          // 38,400,000 floats
#define TOTAL4   (TOTAL / 4)              // 9,600,000 float4
#define NU4      (N_USERS * DIM / 4)      // 3,200,000 float4

typedef float v2f __attribute__((ext_vector_type(2)));
typedef float v8f __attribute__((ext_vector_type(8)));

// ---------------------------------------------------------------------------
// 1) x_cur = concat(user, item);  acc(d_out) = same.
// ---------------------------------------------------------------------------
__global__ void lg_init_x(const float4* __restrict__ uw, const float4* __restrict__ iw,
                          float4* __restrict__ x, float4* __restrict__ acc) {
  int i = blockIdx.x * blockDim.x + threadIdx.x;
  if (i >= TOTAL4) return;
  float4 v = (i < NU4) ? uw[i] : iw[i - NU4];
  x[i] = v;
  acc[i] = v;
}

// ---------------------------------------------------------------------------
// 2) in-degree at dst (deg pre-zeroed with hipMemsetAsync)
// ---------------------------------------------------------------------------
__global__ void lg_degree(const int* __restrict__ edge, int* __restrict__ deg) {
  int e = blockIdx.x * blockDim.x + threadIdx.x;
  if (e >= N_EDGES) return;
  atomicAdd(&deg[edge[N_EDGES + e]], 1);
}

// ---------------------------------------------------------------------------
// 3) exclusive scan of deg -> row_start[0..N_NODES]   (single 1024-thread block)
// ---------------------------------------------------------------------------
__global__ void lg_scan(const int* __restrict__ deg, int* __restrict__ row_start) {
  __shared__ int sdata[1024];
  __shared__ int s_base;
  const int tid = threadIdx.x;
  if (tid == 0) s_base = 0;
  __syncthreads();
  for (int base = 0; base < N_NODES; base += 1024) {
    int i = base + tid;
    int v = (i < N_NODES) ? deg[i] : 0;
    sdata[tid] = v;
    __syncthreads();
    // Hillis-Steele inclusive scan
    for (int off = 1; off < 1024; off <<= 1) {
      int t = (tid >= off) ? sdata[tid - off] : 0;
      __syncthreads();
      sdata[tid] += t;
      __syncthreads();
    }
    int incl = sdata[tid];
    int sb = s_base;
    if (i < N_NODES) row_start[i] = sb + incl - v;   // exclusive
    __syncthreads();
    if (tid == 0) s_base += sdata[1023];
    __syncthreads();
  }
  if (tid == 0) row_start[N_NODES] = s_base;         // == N_EDGES
}

// ---------------------------------------------------------------------------
// 4) dinv = deg>0 ? rsqrt(deg) : 0 ;  cursor = row_start copy
// ---------------------------------------------------------------------------
__global__ void lg_dinv_cursor(const int* __restrict__ deg, const int* __restrict__ row_start,
                               float* __restrict__ dinv, int* __restrict__ cursor) {
  int i = blockIdx.x * blockDim.x + threadIdx.x;
  if (i >= N_NODES) return;
  int d = deg[i];
  dinv[i] = (d > 0) ? rsqrtf((float)d) : 0.0f;
  cursor[i] = row_start[i];
}

// ---------------------------------------------------------------------------
// 5) CSR fill: col[row_start[dst] ...] = src
// ---------------------------------------------------------------------------
__global__ void lg_csr_fill(const int* __restrict__ edge, int* __restrict__ cursor,
                            int* __restrict__ col) {
  int e = blockIdx.x * blockDim.x + threadIdx.x;
  if (e >= N_EDGES) return;
  int s = edge[e];
  int d = edge[N_EDGES + e];
  int pos = atomicAdd(&cursor[d], 1);
  col[pos] = s;
}

// ---------------------------------------------------------------------------
// 6) Pull layer: one wave32 per dst node, float4 per lane (32*4 = 128 dims).
//    x_out[n] = dinv[n] * sum_{s in N(n)} dinv[s] * x_in[s];  acc += x_out[n]
//    No float atomics. Prefetch next gathered row (global_prefetch_b8).
// ---------------------------------------------------------------------------
__global__ void lg_pull(const float4* __restrict__ xin, float4* __restrict__ xout,
                        float4* __restrict__ acc, const int* __restrict__ rs,
                        const int* __restrict__ col, const float* __restrict__ dinv) {
  int wid  = (blockIdx.x * blockDim.x + threadIdx.x) >> 5;
  int lane = threadIdx.x & 31;
  if (wid >= N_NODES) return;
  int s = rs[wid];
  int e = rs[wid + 1];
  float4 sum = make_float4(0.f, 0.f, 0.f, 0.f);
  int src_next = (s < e) ? col[s] : 0;
  for (int j = s; j < e; ++j) {
    int src = src_next;
    if (j + 1 < e) {
      src_next = col[j + 1];
      __builtin_prefetch(&xin[src_next * 32 + lane], 0, 3);
    }
    float  w = dinv[src];
    float4 v = xin[src * 32 + lane];
    sum.x = fmaf(w, v.x, sum.x);
    sum.y = fmaf(w, v.y, sum.y);
    sum.z = fmaf(w, v.z, sum.z);
    sum.w = fmaf(w, v.w, sum.w);
  }
  float dn = dinv[wid];
  float4 r = make_float4(dn * sum.x, dn * sum.y, dn * sum.z, dn * sum.w);
  int o = wid * 32 + lane;
  xout[o] = r;
  float4 a = acc[o];
  a.x += r.x; a.y += r.y; a.z += r.z; a.w += r.w;
  acc[o] = a;
}

// ---------------------------------------------------------------------------
// 7) Epilogue: acc *= 0.25 done as D = A x (0.25*I16) via four chained
//    V_WMMA_F32_16X16X4_F32 steps per 16x16 tile (numerically exact).
//    One wave per 256-float tile; tile is wave-private so in-place is safe.
//    Layouts per CDNA5 ISA 7.12.2:
//      A 16x4 : VGPR0 = K {0 | 2} for lanes {0-15 | 16-31}, VGPR1 = K {1 | 3}
//      B 4x16 : VGPR0 = rows K {0 | 2} striped across lane halves (N = lane&15)
//      C/D    : VGPR v -> M = v (+8 for hi lanes), N = lane&15
// ---------------------------------------------------------------------------
__global__ void lg_final_wmma(float* __restrict__ acc) {
  int wid  = (blockIdx.x * blockDim.x + threadIdx.x) >> 5;
  int lane = threadIdx.x & 31;
  int base = wid * 256;
  if (base >= TOTAL) return;
  const float* t = acc + base;
  const int  m  = lane & 15;
  const bool hi = lane >= 16;

  v8f c = {};
#pragma unroll
  for (int k = 0; k < 4; ++k) {
    int c0 = 4 * k + (hi ? 2 : 0);
    v2f a, b;
    a.x = t[m * 16 + c0];
    a.y = t[m * 16 + c0 + 1];
    b.x = (m == c0)     ? 0.25f : 0.0f;   // B[K=c0,   N=m]
    b.y = (m == c0 + 1) ? 0.25f : 0.0f;   // B[K=c0+1, N=m]
    c = __builtin_amdgcn_wmma_f32_16x16x4_f32(
        /*neg_a=*/false, a, /*neg_b=*/false, b,
        /*c_mod=*/(short)0, c, /*reuse_a=*/false, /*reuse_b=*/false);
  }
#pragma unroll
  for (int v = 0; v < 8; ++v) {
    int mm = hi ? (v + 8) : v;
    acc[base + mm * 16 + m] = c[v];
  }
}

// ---------------------------------------------------------------------------
extern "C" void kernel_launch(void* const* d_in, const int* in_sizes, int n_in,
                              void* d_out, int out_size, void* d_ws, size_t ws_size,
                              hipStream_t stream) {
  const int*   edge = (const int*)d_in[0];     // [2, N_EDGES] int32
  const float* uw   = (const float*)d_in[1];   // [N_USERS, 128]
  const float* iw   = (const float*)d_in[2];   // [N_ITEMS, 128]
  float* acc = (float*)d_out;                  // [N_NODES, 128] accumulator/output

  // workspace carve-up (256B aligned slabs)
  char* ws = (char*)d_ws;
  auto align256 = [](size_t x) { return (x + 255) & ~(size_t)255; };
  float* x_a      = (float*)ws;                ws += align256(sizeof(float) * TOTAL);
  float* x_b      = (float*)ws;                ws += align256(sizeof(float) * TOTAL);
  int*   deg      = (int*)ws;                  ws += align256(sizeof(int) * N_NODES);
  int*   rowst    = (int*)ws;                  ws += align256(sizeof(int) * (N_NODES + 1));
  int*   cursor   = (int*)ws;                  ws += align256(sizeof(int) * N_NODES);
  float* dinv     = (float*)ws;                ws += align256(sizeof(float) * N_NODES);
  int*   col      = (int*)ws;                  ws += align256(sizeof(int) * N_EDGES);

  const int B = 256;

  // init embeddings + accumulator
  lg_init_x<<<(TOTAL4 + B - 1) / B, B, 0, stream>>>(
      (const float4*)uw, (const float4*)iw, (float4*)x_a, (float4*)acc);

  // CSR build
  hipMemsetAsync(deg, 0, sizeof(int) * N_NODES, stream);
  lg_degree<<<(N_EDGES + B - 1) / B, B, 0, stream>>>(edge, deg);
  lg_scan<<<1, 1024, 0, stream>>>(deg, rowst);
  lg_dinv_cursor<<<(N_NODES + B - 1) / B, B, 0, stream>>>(deg, rowst, dinv, cursor);
  lg_csr_fill<<<(N_EDGES + B - 1) / B, B, 0, stream>>>(edge, cursor, col);

  // 3 propagation layers, pull-based (no float atomics), acc fused
  float* xin = x_a;
  float* xout = x_b;
  const int pull_grid = (N_NODES * 32 + B - 1) / B;   // one wave32 per node
  for (int l = 0; l < N_LAYERS; ++l) {
    lg_pull<<<pull_grid, B, 0, stream>>>(
        (const float4*)xin, (float4*)xout, (float4*)acc, rowst, col, dinv);
    float* tmp = xin; xin = xout; xout = tmp;
  }

  // mean pooling (/4) through the f32 WMMA pipe
  const int wmma_grid = (TOTAL / 256) / (B / 32);     // 150000 waves / 8 per block
  lg_final_wmma<<<wmma_grid, B, 0, stream>>>(acc);
}